// Aggre_81226421502275
// MI455X (gfx1250) — compile-verified
//
#include <hip/hip_runtime.h>

// MI455X (gfx1250) kernel: per-(node,neighbor) MLP 7->40->3 + mean over 16 neighbors.
//   y[n] = mean_k( relu(mailbox[n,k,:] @ W1 + b1) ) @ W2 + b2   (layer-2 hoisted out of mean)
// One node == one V_WMMA_F32_16X16X4_F32 tile set: A = 16x8 (messages x padded features),
// B = W1 8x16 column-tile (3 tiles cover 40 hidden units), 2 k-steps of 4.
// b1 folded into loop-invariant C fragments; ReLU via 0.5*(x+|x|) (|x| is a free source
// modifier, bit-exact here); the 0.5 and the 1/16 neighbor-mean are folded into W2.

typedef __attribute__((ext_vector_type(2))) float v2f;
typedef __attribute__((ext_vector_type(8))) float v8f;

#define KNB  16
#define FIN  7
#define FHID 40
#define FOUT 3

__global__ __launch_bounds__(256) void mailbox_mlp_wmma(
    const float* __restrict__ mailbox,  // [N,16,7]
    const float* __restrict__ W1,       // [7,40]
    const float* __restrict__ b1,       // [40]
    const float* __restrict__ W2,       // [40,3]
    const float* __restrict__ b2,       // [3]
    float* __restrict__ out,            // [N,3]
    int n_nodes)
{
    const int lane         = threadIdx.x & 31;
    const int waveInBlock  = threadIdx.x >> 5;
    const int wavesPerBlk  = blockDim.x >> 5;
    const int waveId       = blockIdx.x * wavesPerBlk + waveInBlock;
    const int numWaves     = gridDim.x * wavesPerBlk;

    const int  colLo  = lane & 15;      // 0..15: column within tile / message row
    const bool hiHalf = lane >= 16;

    // ---- Hoist weights into registers (per wave, reused across all nodes) ----
    // B fragment layout for 16x16x4 f32 (4x16): lanes 0-15 col=lane {V0=K0,V1=K1},
    // lanes 16-31 col=lane-16 {V0=K2,V1=K3}  (mirrors documented A/C striping).
    v2f   Bfrag[3][2];
    float b1v[3];
    float w2v[3][3];
#pragma unroll
    for (int t = 0; t < 3; ++t) {
        const int  col = t * 16 + colLo;
        const bool cv  = (col < FHID);
#pragma unroll
        for (int ks = 0; ks < 2; ++ks) {
            const int k0 = ks * 4 + (hiHalf ? 2 : 0);     // feature row of W1
            float x0 = (cv && (k0     < FIN)) ? W1[ k0      * FHID + col] : 0.0f;
            float x1 = (cv && (k0 + 1 < FIN)) ? W1[(k0 + 1) * FHID + col] : 0.0f;
            Bfrag[t][ks].x = x0;
            Bfrag[t][ks].y = x1;
        }
        b1v[t] = cv ? b1[col] : 0.0f;
        // fold 0.5 (ReLU identity) * 1/16 (neighbor mean) into the W2 copies
#pragma unroll
        for (int o = 0; o < FOUT; ++o)
            w2v[t][o] = cv ? (W2[col * FOUT + o] * (0.5f / (float)KNB)) : 0.0f;
    }
    const float b2v0 = b2[0], b2v1 = b2[1], b2v2 = b2[2];

    // A fragment addressing: row (message) = colLo.
    // k-step 0: lanes 0-15 features {0,1}, lanes 16-31 features {2,3}
    // k-step 1: lanes 0-15 features {4,5}, lanes 16-31 features {6, pad0}
    const int row  = colLo;
    const int off0 = row * FIN + (hiHalf ? 2 : 0);
    const int off1 = row * FIN + (hiHalf ? 6 : 4);
    const int offY = row * FIN + 5;     // in-bounds for ALL lanes; hi half discards it

    for (int n = waveId; n < n_nodes; n += numWaves) {
        const float* mb = mailbox + (size_t)n * (KNB * FIN);

        // Branch-free prefetch of the next node's tile: clamp index with cndmask so the
        // address is always in-bounds (no EXEC round-trip in the hot loop).
        {
            const int pn = (n + numWaves < n_nodes) ? (n + numWaves) : n;
            __builtin_prefetch(mailbox + (size_t)pn * (KNB * FIN) + off0, 0, 3);
        }

        // Unconditional loads (no EXEC divergence); zero the K=3 pad via cndmask.
        const float a0x = mb[off0];
        const float a0y = mb[off0 + 1];
        const float a1x = mb[off1];
        const float eY  = mb[offY];
        v2f A0, A1;
        A0.x = a0x;
        A0.y = a0y;
        A1.x = a1x;
        A1.y = hiHalf ? 0.0f : eY;      // feature 7 is K-padding (B row is zero too)

        float acc0 = 0.0f, acc1 = 0.0f, acc2 = 0.0f;
#pragma unroll
        for (int t = 0; t < 3; ++t) {
            v8f c;
#pragma unroll
            for (int j = 0; j < 8; ++j) c[j] = b1v[t];   // loop-invariant; hoisted by LLVM
            c = __builtin_amdgcn_wmma_f32_16x16x4_f32(false, A0, false, Bfrag[t][0],
                                                      (short)0, c, false, false);
            c = __builtin_amdgcn_wmma_f32_16x16x4_f32(false, A1, false, Bfrag[t][1],
                                                      (short)0, c, false, false);
            // 2*ReLU row-sum: this lane holds 8 rows of one hidden column.
            // relu(x)*2 == x + |x| exactly; |x| is a free VALU source modifier.
            float s = 0.0f;
#pragma unroll
            for (int j = 0; j < 8; ++j) s += (c[j] + __builtin_fabsf(c[j]));
            acc0 = fmaf(s, w2v[t][0], acc0);
            acc1 = fmaf(s, w2v[t][1], acc1);
            acc2 = fmaf(s, w2v[t][2], acc2);
        }
        // Wave32 all-lane reduction: sums rows (partner half-lanes) and hidden columns.
#pragma unroll
        for (int d = 16; d >= 1; d >>= 1) {
            acc0 += __shfl_xor(acc0, d, 32);
            acc1 += __shfl_xor(acc1, d, 32);
            acc2 += __shfl_xor(acc2, d, 32);
        }
        if (lane == 0) {
            float* po = out + (size_t)n * FOUT;
            po[0] = acc0 + b2v0;
            po[1] = acc1 + b2v1;
            po[2] = acc2 + b2v2;
        }
    }
}

extern "C" void kernel_launch(void* const* d_in, const int* in_sizes, int n_in,
                              void* d_out, int out_size, void* d_ws, size_t ws_size,
                              hipStream_t stream) {
    const float* mailbox = (const float*)d_in[0];
    const float* W1      = (const float*)d_in[1];
    const float* b1      = (const float*)d_in[2];
    const float* W2      = (const float*)d_in[3];
    const float* b2      = (const float*)d_in[4];
    float* out = (float*)d_out;

    const int n_nodes = in_sizes[0] / (KNB * FIN);   // 500000

    // 2048 blocks x 8 waves = 16384 waves; ~31 nodes per wave (grid-stride).
    const int threads = 256;
    int blocks = 2048;
    const int maxBlocks = (n_nodes + (threads / 32) - 1) / (threads / 32);
    if (blocks > maxBlocks) blocks = maxBlocks;
    if (blocks < 1) blocks = 1;

    mailbox_mlp_wmma<<<blocks, threads, 0, stream>>>(mailbox, W1, b1, W2, b2, out, n_nodes);
}